// MinkOccupancyForecastingNetwork_53214644798122
// MI455X (gfx1250) — compile-verified
//
#include <hip/hip_runtime.h>

// ---------------------------------------------------------------------------
// Volumetric ray marching (dvr.render) for MI455X / gfx1250, wave32.
//   Kernel 1: one wave per ray. Lane L owns steps [L*24, L*24+24).
//             24 UNCONDITIONAL clamped L2 gathers in flight per lane (flat
//             control flow, GVS addressing: SGPR base + 32-bit VGPR offset),
//             local prefix, wave exclusive scan (ds_bpermute), multiplicative
//             transmittance, per-ray loss triple -> workspace.
//   Kernel 2: single fully-active wave reduces the three 10240-length loss
//             arrays with V_WMMA_F32_16X16X4_F32 (A = ones): exact fp32 sum
//             on the matrix unit; total = (sum D)/16; /N_RAYS -> d_out[0..2].
// ---------------------------------------------------------------------------

#define T_DIM   5
#define Z_DIM   32
#define Y_DIM   512
#define X_DIM   512
#define NRAYS   10000
#define NSTEPS  768
#define SPL     (NSTEPS / 32)      // 24 steps per lane
#define RAYS_PAD 10240             // multiple of 64 for WMMA chunks
#define WPB     8                  // waves (rays) per block
#define VOXEL_F 0.2f

typedef float v2f __attribute__((ext_vector_type(2)));
typedef float v8f __attribute__((ext_vector_type(8)));

__global__ __launch_bounds__(256) void ray_march_kernel(
    const float* __restrict__ grid,      // [T,Z,Y,X] raw features (relu applied here)
    const float* __restrict__ origin,    // [T,3]
    const float* __restrict__ points,    // [NRAYS,3]
    const int*   __restrict__ tindex,    // [NRAYS]
    float*       __restrict__ ws)        // 3 x RAYS_PAD floats
{
    const int wave = threadIdx.x >> 5;
    const int lane = threadIdx.x & 31;
    const int ray  = blockIdx.x * WPB + wave;
    if (ray >= RAYS_PAD) return;

    float* l1w = ws;
    float* l2w = ws + RAYS_PAD;
    float* arw = ws + 2 * RAYS_PAD;

    if (ray >= NRAYS) {               // zero the padding lanes of the workspace
        if (lane == 0) { l1w[ray] = 0.f; l2w[ray] = 0.f; arw[ray] = 0.f; }
        return;
    }

    const int   ti      = tindex[ray];
    const float inv_vox = 1.0f / VOXEL_F;
    // voxel-space origin and target (offset = PC_RANGE[:3] = (-51.2,-51.2,-3.2))
    const float ox = (origin[ti * 3 + 0] + 51.2f) * inv_vox;
    const float oy = (origin[ti * 3 + 1] + 51.2f) * inv_vox;
    const float oz = (origin[ti * 3 + 2] +  3.2f) * inv_vox;
    const float qx = (points[ray * 3 + 0] + 51.2f) * inv_vox;
    const float qy = (points[ray * 3 + 1] + 51.2f) * inv_vox;
    const float qz = (points[ray * 3 + 2] +  3.2f) * inv_vox;

    const float dx = qx - ox, dy = qy - oy, dz = qz - oz;
    const float gt = sqrtf(dx * dx + dy * dy + dz * dz);
    const float rinv = 1.0f / fmaxf(gt, 1e-6f);
    const float ux = dx * rinv, uy = dy * rinv, uz = dz * rinv;

    // full element index fits in 32 bits: 5*32*512*512 = 41.9M elements
    const unsigned sliceBase = (unsigned)ti * (unsigned)(Z_DIM * Y_DIM * X_DIM);

    // ---- phase 1: 24 independent clamped gathers per lane (flat flow) ----
    float tau[SPL];
    const int s0 = lane * SPL;
#pragma unroll
    for (int j = 0; j < SPL; ++j) {
        const float t  = (float)(s0 + j) + 0.5f;
        const float fx = ox + ux * t;
        const float fy = oy + uy * t;
        const float fz = oz + uz * t;
        const int ix = (int)floorf(fx);
        const int iy = (int)floorf(fy);
        const int iz = (int)floorf(fz);
        const bool inb = (ix >= 0) & (ix < X_DIM) &
                         (iy >= 0) & (iy < Y_DIM) &
                         (iz >= 0) & (iz < Z_DIM);
        const int ixc = min(max(ix, 0), X_DIM - 1);
        const int iyc = min(max(iy, 0), Y_DIM - 1);
        const int izc = min(max(iz, 0), Z_DIM - 1);
        const unsigned idx = sliceBase +
            (((unsigned)izc * Y_DIM + (unsigned)iyc) * X_DIM + (unsigned)ixc);
        const float s = grid[idx];               // unconditional: stays in a clause
        tau[j] = inb ? fmaxf(s, 0.0f) : 0.0f;    // relu + OOB mask
    }

    // ---- phase 2: chunk sums + wave32 exclusive scan of optical depth ----
    float local = 0.0f;
#pragma unroll
    for (int j = 0; j < SPL; ++j) local += tau[j];

    float incl = local;
#pragma unroll
    for (int off = 1; off < 32; off <<= 1) {
        const float v = __shfl_up(incl, off, 32);
        if (lane >= off) incl += v;
    }
    const float excl = incl - local;  // optical depth before this lane's chunk

    // ---- phase 3: transmittance-weighted expected distance ----
    float trans = expf(-excl);        // exp(-(cumsum - tau)) at chunk start
    float pred  = 0.0f;
#pragma unroll
    for (int j = 0; j < SPL; ++j) {
        const float e = expf(-tau[j]);
        const float t = (float)(s0 + j) + 0.5f;
        pred += trans * (1.0f - e) * t;
        trans *= e;
    }
#pragma unroll
    for (int off = 16; off; off >>= 1) pred += __shfl_xor(pred, off, 32);

    if (lane == 0) {
        const float predm = pred * VOXEL_F;
        const float gtm   = gt * VOXEL_F;
        const bool  valid = (gtm >= 0.0f);
        const float diff  = gtm - predm;
        const float ad    = fabsf(diff);
        l1w[ray] = valid ? ad : 0.0f;
        l2w[ray] = valid ? 0.5f * diff * diff : 0.0f;
        arw[ray] = valid ? ad / fmaxf(gtm, 1e-6f) : 0.0f;
    }
}

// ---------------------------------------------------------------------------
// Final reduction: one fully-active wave (EXEC all ones -> WMMA legal).
// sum(x[0..RAYS_PAD)) via D = ones(16x4) x B(4x16) + C accumulation:
// every element of B contributes to all 16 rows of D, so total = (sum D)/16.
// ---------------------------------------------------------------------------
__global__ __launch_bounds__(32) void loss_reduce_kernel(
    const float* __restrict__ ws, float* __restrict__ out)
{
    const int   lane  = threadIdx.x;
    const float scale = 1.0f / (float)NRAYS;   // count == NRAYS (gt >= 0 always)

#if __has_builtin(__builtin_amdgcn_wmma_f32_16x16x4_f32)
    v2f ones; ones[0] = 1.0f; ones[1] = 1.0f;
    for (int ch = 0; ch < 3; ++ch) {
        const float* __restrict__ p = ws + ch * RAYS_PAD;
        v8f c = {0.f, 0.f, 0.f, 0.f, 0.f, 0.f, 0.f, 0.f};
        for (int base = 0; base < RAYS_PAD; base += 64) {
            v2f b;
            b[0] = p[base + lane];
            b[1] = p[base + 32 + lane];
            c = __builtin_amdgcn_wmma_f32_16x16x4_f32(
                    false, ones, false, b, (short)0, c, false, false);
        }
        float s = 0.0f;
#pragma unroll
        for (int v = 0; v < 8; ++v) s += c[v];
#pragma unroll
        for (int off = 16; off; off >>= 1) s += __shfl_xor(s, off, 32);
        if (lane == 0) out[ch] = s * (1.0f / 16.0f) * scale;
    }
#else
    for (int ch = 0; ch < 3; ++ch) {
        const float* __restrict__ p = ws + ch * RAYS_PAD;
        float s = 0.0f;
        for (int i = lane; i < RAYS_PAD; i += 32) s += p[i];
#pragma unroll
        for (int off = 16; off; off >>= 1) s += __shfl_xor(s, off, 32);
        if (lane == 0) out[ch] = s * scale;
    }
#endif
}

extern "C" void kernel_launch(void* const* d_in, const int* in_sizes, int n_in,
                              void* d_out, int out_size, void* d_ws, size_t ws_size,
                              hipStream_t stream)
{
    const float* grid   = (const float*)d_in[0]; // [1,5,32,512,512] f32
    const float* origin = (const float*)d_in[1]; // [1,5,3]          f32
    const float* points = (const float*)d_in[2]; // [1,10000,3]      f32
    const int*   tindex = (const int*)  d_in[3]; // [1,10000]        i32
    float*       ws     = (float*)d_ws;          // 3 * RAYS_PAD floats
    float*       out    = (float*)d_out;         // 3 floats [l1,l2,absrel]

    (void)in_sizes; (void)n_in; (void)out_size; (void)ws_size;

    const dim3 block(256);                               // 8 waves = 8 rays
    const dim3 gridDim((RAYS_PAD + WPB - 1) / WPB);      // 1280 blocks
    ray_march_kernel<<<gridDim, block, 0, stream>>>(grid, origin, points, tindex, ws);
    loss_reduce_kernel<<<1, 32, 0, stream>>>(ws, out);
}